// RotarySelfAttention_55637006352457
// MI455X (gfx1250) — compile-verified
//
#include <hip/hip_runtime.h>
#include <math.h>

// ---------------- problem constants ----------------
#define D_MODEL   2048
#define NUM_HEADS 16
#define NUM_KV    4
#define HEAD_DIM  128
#define GROUPS    4
#define ROT_DIM   64
#define BATCH     2
#define SEQ       2048
#define BL        (BATCH * SEQ)   // 4096 rows

typedef __attribute__((ext_vector_type(2))) float v2f;
typedef __attribute__((ext_vector_type(8))) float v8f;
typedef __attribute__((ext_vector_type(4))) unsigned int u32x4;
typedef __attribute__((ext_vector_type(4))) int i32x4;
typedef __attribute__((ext_vector_type(8))) int i32x8;

#if defined(__has_builtin)
#  if __has_builtin(__builtin_amdgcn_tensor_load_to_lds) && \
      __has_builtin(__builtin_amdgcn_s_wait_tensorcnt)
#    define USE_TDM 1
#  endif
#endif
#ifndef USE_TDM
#  define USE_TDM 0
#endif

static __device__ __forceinline__ v8f wmma4(v2f a, v2f b, v8f c) {
  // D = A(16x4) * B(4x16) + C(16x16), fp32 matrix pipe
  return __builtin_amdgcn_wmma_f32_16x16x4_f32(false, a, false, b, (short)0, c,
                                               false, false);
}

#if USE_TDM
// Build a 2D-tile Tensor DMA descriptor (D# groups 0/1, ISA ch.8).
// dims/strides in elements (data_size = 4 bytes). LDS padding applied by TDM:
// pad_int_enc: pad after 2^(enc+1) DWORDs; pad_amt_enc: insert (enc+1) DWORDs.
static __device__ __forceinline__ void tdm_make(
    u32x4& g0, i32x8& g1, const void* gaddr, unsigned lds_addr,
    unsigned tensor_d0, unsigned tensor_d1, unsigned tile_d0, unsigned tile_d1,
    unsigned long long stride0, unsigned pad_int_enc, unsigned pad_amt_enc)
{
  const unsigned long long ga = (unsigned long long)gaddr;
  g0.x = 1u;                                            // count=1, user mode
  g0.y = lds_addr;                                      // LDS byte address
  g0.z = (unsigned)(ga & 0xffffffffu);                  // global_addr[31:0]
  g0.w = (unsigned)((ga >> 32) & 0x1ffffffu) | (2u << 30);  // [56:32] | type=2

  unsigned long long w0 = 0, w1 = 0, w2 = 0, w3 = 0;
  w0 |= (2ull << 16);                                   // data_size = 4B
  w0 |= (1ull << 20);                                   // pad_enable
  w0 |= ((unsigned long long)(pad_int_enc & 7u)) << 22;
  w0 |= ((unsigned long long)(pad_amt_enc & 127u)) << 25;
  w0 |= ((unsigned long long)(tensor_d0 & 0xffffu)) << 48;   // tensor_dim0 lo
  w1 |= (unsigned long long)(tensor_d0 >> 16);               // tensor_dim0 hi
  w1 |= ((unsigned long long)tensor_d1) << 16;               // tensor_dim1
  w1 |= ((unsigned long long)(tile_d0 & 0xffffu)) << 48;     // tile_dim0
  w2 |= (unsigned long long)(tile_d1 & 0xffffu);             // tile_dim1
  w2 |= (stride0 & 0xffffffffull) << 32;                     // stride0[31:0]
  w3 |= (stride0 >> 32) & 0xffffull;                         // stride0[47:32]
  // tile_dim2 = 0 (2D), tensor_dim1_stride = 0, wg_mask = 0 (not in cluster)
  g1[0] = (int)(w0 & 0xffffffffull); g1[1] = (int)(w0 >> 32);
  g1[2] = (int)(w1 & 0xffffffffull); g1[3] = (int)(w1 >> 32);
  g1[4] = (int)(w2 & 0xffffffffull); g1[5] = (int)(w2 >> 32);
  g1[6] = (int)(w3 & 0xffffffffull); g1[7] = (int)(w3 >> 32);
}

static __device__ __forceinline__ void tdm_issue(u32x4 g0, i32x8 g1) {
  const i32x4 z4 = (i32x4){0, 0, 0, 0};
#if defined(__clang_major__) && (__clang_major__ >= 23)
  const i32x8 z8 = (i32x8){0, 0, 0, 0, 0, 0, 0, 0};
  __builtin_amdgcn_tensor_load_to_lds(g0, g1, z4, z4, z8, 0);
#else
  __builtin_amdgcn_tensor_load_to_lds(g0, g1, z4, z4, 0);
#endif
}
#endif  // USE_TDM

// =====================================================================
// GEMM: C[M,N] = A[M,K] @ B[K,N], all row-major fp32.
// Block tile 128x128, 256 threads = 8 waves in a 4(M) x 2(N) grid;
// each wave computes 32x64 via 2x4 WMMA 16x16 accumulators.
// K stepped by 16. Tiles staged into LDS by the Tensor Data Mover,
// double-buffered so the DMA for chunk i+1 overlaps WMMAs on chunk i.
// TDM inserts the 4-DWORD row padding (As rows 16+4, Bs rows 128+4).
// Requires M%128==0, N%128==0, K%16==0 (true for all calls here).
// =====================================================================
__global__ __launch_bounds__(256)
void gemm_f32(const float* __restrict__ A, const float* __restrict__ B,
              float* __restrict__ C, int M, int N, int K)
{
  __shared__ float As[2][128][20];   // 128 rows x 16 K (+4 pad)
  __shared__ float Bs[2][16][132];   // 16 K x 128 cols (+4 pad)

  const int tid  = threadIdx.x;
  const int lane = tid & 31;
  const int wave = tid >> 5;
  const int half = lane >> 4;     // which 16-lane half
  const int l16  = lane & 15;

  const int bm = blockIdx.x * 128;
  const int bn = blockIdx.y * 128;
  const int wm = (wave >> 1) * 32;   // wave M offset: 0,32,64,96
  const int wn = (wave & 1) * 64;    // wave N offset: 0,64

  v8f acc[2][4];
#pragma unroll
  for (int i = 0; i < 2; ++i)
#pragma unroll
    for (int j = 0; j < 4; ++j)
      acc[i][j] = (v8f){0.f,0.f,0.f,0.f,0.f,0.f,0.f,0.f};

  const int nchunks = K >> 4;

#if USE_TDM
  // ---- wave 0 drives the Tensor Data Mover; others just compute ----
  if (wave == 0) {
    u32x4 g0; i32x8 g1;
    // chunk 0 -> buffer 0
    tdm_make(g0, g1, A + (size_t)bm * K, (unsigned)(uintptr_t)&As[0][0][0],
             (unsigned)K, 128u, 16u, 128u, (unsigned long long)K,
             /*pad 16 dw*/3u, /*+4 dw*/3u);
    tdm_issue(g0, g1);
    tdm_make(g0, g1, B + bn, (unsigned)(uintptr_t)&Bs[0][0][0],
             (unsigned)N, (unsigned)K, 128u, 16u, (unsigned long long)N,
             /*pad 128 dw*/6u, /*+4 dw*/3u);
    tdm_issue(g0, g1);
  }

  for (int i = 0; i < nchunks; ++i) {
    const int buf = i & 1;
    if (wave == 0) {
      if (i + 1 < nchunks) {
        const int k1 = (i + 1) << 4;
        u32x4 g0; i32x8 g1;
        tdm_make(g0, g1, A + (size_t)bm * K + k1,
                 (unsigned)(uintptr_t)&As[buf ^ 1][0][0],
                 (unsigned)K, 128u, 16u, 128u, (unsigned long long)K, 3u, 3u);
        tdm_issue(g0, g1);
        tdm_make(g0, g1, B + (size_t)k1 * N + bn,
                 (unsigned)(uintptr_t)&Bs[buf ^ 1][0][0],
                 (unsigned)N, (unsigned)K, 128u, 16u, (unsigned long long)N,
                 6u, 3u);
        tdm_issue(g0, g1);
        __builtin_amdgcn_s_wait_tensorcnt(2);  // 2 in flight -> chunk i done
      } else {
        __builtin_amdgcn_s_wait_tensorcnt(0);
      }
    }
    __syncthreads();   // chunk i visible to all waves

#pragma unroll
    for (int s = 0; s < 4; ++s) {
      const int kk = 4 * s + 2 * half;
      v2f af[2], bf[4];
#pragma unroll
      for (int mt = 0; mt < 2; ++mt) {
        const int r = wm + mt * 16 + l16;
        af[mt].x = As[buf][r][kk];
        af[mt].y = As[buf][r][kk + 1];
      }
#pragma unroll
      for (int nt = 0; nt < 4; ++nt) {
        const int cc = wn + nt * 16 + l16;
        bf[nt].x = Bs[buf][kk][cc];
        bf[nt].y = Bs[buf][kk + 1][cc];
      }
#pragma unroll
      for (int mt = 0; mt < 2; ++mt)
#pragma unroll
        for (int nt = 0; nt < 4; ++nt)
          acc[mt][nt] = wmma4(af[mt], bf[nt], acc[mt][nt]);
    }
    __syncthreads();   // all readers done before TDM reuses this buffer
  }
#else  // !USE_TDM: cooperative float4 staging fallback
  const int arow = tid >> 1;
  const int acol = (tid & 1) * 8;
  const int brow = tid >> 4;
  const int bcol = (tid & 15) * 8;
  for (int i = 0; i < nchunks; ++i) {
    const int k0 = i << 4;
    const int buf = i & 1;
    const float4* ag = (const float4*)(A + (size_t)(bm + arow) * K + k0 + acol);
    *(float4*)&As[buf][arow][acol]     = ag[0];
    *(float4*)&As[buf][arow][acol + 4] = ag[1];
    const float4* bg = (const float4*)(B + (size_t)(k0 + brow) * N + bn + bcol);
    *(float4*)&Bs[buf][brow][bcol]     = bg[0];
    *(float4*)&Bs[buf][brow][bcol + 4] = bg[1];
    __syncthreads();
#pragma unroll
    for (int s = 0; s < 4; ++s) {
      const int kk = 4 * s + 2 * half;
      v2f af[2], bf[4];
#pragma unroll
      for (int mt = 0; mt < 2; ++mt) {
        const int r = wm + mt * 16 + l16;
        af[mt].x = As[buf][r][kk];
        af[mt].y = As[buf][r][kk + 1];
      }
#pragma unroll
      for (int nt = 0; nt < 4; ++nt) {
        const int cc = wn + nt * 16 + l16;
        bf[nt].x = Bs[buf][kk][cc];
        bf[nt].y = Bs[buf][kk + 1][cc];
      }
#pragma unroll
      for (int mt = 0; mt < 2; ++mt)
#pragma unroll
        for (int nt = 0; nt < 4; ++nt)
          acc[mt][nt] = wmma4(af[mt], bf[nt], acc[mt][nt]);
    }
    __syncthreads();
  }
#endif

  // ---- epilogue: C/D layout = VGPR r -> row r (+8 for high half), col = l16
#pragma unroll
  for (int mt = 0; mt < 2; ++mt)
#pragma unroll
    for (int nt = 0; nt < 4; ++nt)
#pragma unroll
      for (int r = 0; r < 8; ++r) {
        const int row = bm + wm + mt * 16 + r + 8 * half;
        const int col = bn + wn + nt * 16 + l16;
        C[(size_t)row * N + col] = acc[mt][nt][r];
      }
}

// =====================================================================
// RoPE + RMS-norm, in place. One wave per (b,l,head) row of 128 floats.
// Lane owns dims {l, l+32} paired with {l+64, l+96}.
// =====================================================================
__global__ __launch_bounds__(256)
void rope_rms(float* __restrict__ t, const float* __restrict__ cs,
              const float* __restrict__ sn, int nheads, int nrows)
{
  const int lane = threadIdx.x & 31;
  const int wid  = blockIdx.x * 8 + (threadIdx.x >> 5);
  if (wid >= nrows) return;
  const int l = (wid / nheads) % SEQ;

  float* row = t + (size_t)wid * HEAD_DIM;
  const float* cr = cs + (size_t)l * ROT_DIM;
  const float* sr = sn + (size_t)l * ROT_DIM;

  const int i1 = lane, i2 = lane + 32;
  const float c1 = cr[i1], s1 = sr[i1];
  const float c2 = cr[i2], s2 = sr[i2];
  const float a1 = row[i1],           b1 = row[i1 + ROT_DIM];
  const float a2 = row[i2],           b2 = row[i2 + ROT_DIM];
  const float o0 = a1 * c1 - b1 * s1;           // dim i1
  const float o1 = a2 * c2 - b2 * s2;           // dim i2
  const float o2 = b1 * c1 + a1 * s1;           // dim i1+64
  const float o3 = b2 * c2 + a2 * s2;           // dim i2+64

  float ss = o0 * o0 + o1 * o1 + o2 * o2 + o3 * o3;
  ss += __shfl_xor(ss, 16, 32);
  ss += __shfl_xor(ss,  8, 32);
  ss += __shfl_xor(ss,  4, 32);
  ss += __shfl_xor(ss,  2, 32);
  ss += __shfl_xor(ss,  1, 32);
  const float inv = rsqrtf(ss * (1.0f / 128.0f) + 1e-6f);

  row[i1]           = o0 * inv;
  row[i2]           = o1 * inv;
  row[i1 + ROT_DIM] = o2 * inv;
  row[i2 + ROT_DIM] = o3 * inv;
}

// =====================================================================
// Causal GQA flash attention. One wave per (b, head, 16-query tile).
// S = Q*K^T via 32 fp32 WMMAs (K-dim 128); online softmax on the 16x16
// C-layout tile (row stats reduced across 16-lane halves); P re-shaped
// through LDS into A-layout; P*V via 4x8 WMMAs into 8 v8f accumulators.
// Next K/V tiles prefetched (global_prefetch_b8) behind current WMMAs.
// =====================================================================
__global__ __launch_bounds__(32)
void flash_attn(const float* __restrict__ q, const float* __restrict__ k,
                const float* __restrict__ v, float* __restrict__ o)
{
  __shared__ float plds[16][17];
  const int lane = threadIdx.x & 31;
  const int half = lane >> 4;
  const int l16  = lane & 15;

  int idx = blockIdx.x;
  const int qt  = idx & (SEQ / 16 - 1);   // 0..127
  idx >>= 7;
  const int h   = idx & (NUM_HEADS - 1);
  const int b   = idx >> 4;
  const int kvh = h >> 2;                 // GQA: 4 Q heads per KV head

  const float scale = 0.08838834764831845f;  // 1/sqrt(128), folded into Q

  // ---- preload Q tile (16 x 128) as 32 A-fragments ----
  const int qrow = qt * 16 + l16;
  const float* qp = q + ((size_t)(b * SEQ + qrow) * NUM_HEADS + h) * HEAD_DIM;
  v2f qa[32];
#pragma unroll
  for (int s = 0; s < 32; ++s) {
    const int kk = 4 * s + 2 * half;
    v2f tmp = *(const v2f*)(qp + kk);
    qa[s] = tmp * scale;
  }

  v8f oacc[8];
#pragma unroll
  for (int j = 0; j < 8; ++j) oacc[j] = (v8f){0.f,0.f,0.f,0.f,0.f,0.f,0.f,0.f};
  float mrow[8], lrow[8];
#pragma unroll
  for (int r = 0; r < 8; ++r) { mrow[r] = -3.0e38f; lrow[r] = 0.f; }

  const size_t kv_ls = (size_t)NUM_KV * HEAD_DIM;  // 512 floats per l
  const float* kb = k + ((size_t)b * SEQ * NUM_KV + kvh) * HEAD_DIM;
  const float* vb = v + ((size_t)b * SEQ * NUM_KV + kvh) * HEAD_DIM;

  for (int kt = 0; kt <= qt; ++kt) {
    // prefetch next tile's K/V rows while this tile computes
    if (kt + 1 <= qt) {
      const size_t noff = (size_t)((kt + 1) * 16 + l16) * kv_ls;
      __builtin_prefetch(kb + noff, 0, 1);
      __builtin_prefetch(vb + noff, 0, 1);
    }

    // ---- S tile = Q @ K^T : B-frag loads mirror A pattern off key rows ----
    const float* kp = kb + (size_t)(kt * 16 + l16) * kv_ls;
    v8f sacc = (v8f){0.f,0.f,0.f,0.f,0.f,0.f,0.f,0.f};
#pragma unroll
    for (int s = 0; s < 32; ++s) {
      const int kk = 4 * s + 2 * half;
      v2f bf = *(const v2f*)(kp + kk);
      sacc = wmma4(qa[s], bf, sacc);
    }

    if (kt == qt) {  // causal mask on diagonal tile
#pragma unroll
      for (int r = 0; r < 8; ++r)
        if (l16 > r + 8 * half) sacc[r] = -1.0e9f;
    }

    // ---- online softmax (rows live per-half across 16 lanes) ----
#pragma unroll
    for (int r = 0; r < 8; ++r) {
      float sm = sacc[r];
      sm = fmaxf(sm, __shfl_xor(sm, 8, 16));
      sm = fmaxf(sm, __shfl_xor(sm, 4, 16));
      sm = fmaxf(sm, __shfl_xor(sm, 2, 16));
      sm = fmaxf(sm, __shfl_xor(sm, 1, 16));
      const float mnew = fmaxf(mrow[r], sm);
      const float p    = __expf(sacc[r] - mnew);
      float ps = p;
      ps += __shfl_xor(ps, 8, 16);
      ps += __shfl_xor(ps, 4, 16);
      ps += __shfl_xor(ps, 2, 16);
      ps += __shfl_xor(ps, 1, 16);
      const float alpha = __expf(mrow[r] - mnew);
      lrow[r] = lrow[r] * alpha + ps;
      mrow[r] = mnew;
#pragma unroll
      for (int j = 0; j < 8; ++j) oacc[j][r] *= alpha;
      plds[r + 8 * half][l16] = p;   // C-layout -> LDS (row = M, col = key)
    }
    __syncthreads();  // single-wave WG (S_NOP); LDS DS ops are in-order

    // ---- oacc += P @ V ----
    const float* vp = vb + (size_t)(kt * 16) * kv_ls;
#pragma unroll
    for (int s = 0; s < 4; ++s) {
      const int kk = 4 * s + 2 * half;
      v2f pa;
      pa.x = plds[l16][kk];
      pa.y = plds[l16][kk + 1];
#pragma unroll
      for (int j = 0; j < 8; ++j) {
        v2f bf;
        bf.x = vp[(size_t)kk * kv_ls + j * 16 + l16];
        bf.y = vp[(size_t)(kk + 1) * kv_ls + j * 16 + l16];
        oacc[j] = wmma4(pa, bf, oacc[j]);
      }
    }
    __syncthreads();
  }

  // ---- normalize + write out [B,L,H,D] ----
  float* op = o + ((size_t)(b * SEQ + qt * 16) * NUM_HEADS + h) * HEAD_DIM;
#pragma unroll
  for (int r = 0; r < 8; ++r) {
    const float inv = 1.0f / lrow[r];
    const int m = r + 8 * half;
#pragma unroll
    for (int j = 0; j < 8; ++j)
      op[(size_t)m * NUM_HEADS * HEAD_DIM + j * 16 + l16] = oacc[j][r] * inv;
  }
}

// =====================================================================
extern "C" void kernel_launch(void* const* d_in, const int* in_sizes, int n_in,
                              void* d_out, int out_size, void* d_ws, size_t ws_size,
                              hipStream_t stream)
{
  const float* x  = (const float*)d_in[0];
  const float* cs = (const float*)d_in[1];
  const float* sn = (const float*)d_in[2];
  const float* Wq = (const float*)d_in[3];
  const float* Wk = (const float*)d_in[4];
  const float* Wv = (const float*)d_in[5];
  const float* Wo = (const float*)d_in[6];
  float* out = (float*)d_out;

  // workspace: q (32MB) | k (8MB) | v (8MB) | attn_out (32MB) = 80MB
  float* q_ws = (float*)d_ws;
  float* k_ws = q_ws + (size_t)BL * D_MODEL;
  float* v_ws = k_ws + (size_t)BL * NUM_KV * HEAD_DIM;
  float* a_ws = v_ws + (size_t)BL * NUM_KV * HEAD_DIM;

  const dim3 blk(256);
  const int KVN = NUM_KV * HEAD_DIM;  // 512

  gemm_f32<<<dim3(BL / 128, D_MODEL / 128), blk, 0, stream>>>(
      x, Wq, q_ws, BL, D_MODEL, D_MODEL);
  gemm_f32<<<dim3(BL / 128, KVN / 128), blk, 0, stream>>>(
      x, Wk, k_ws, BL, KVN, D_MODEL);
  gemm_f32<<<dim3(BL / 128, KVN / 128), blk, 0, stream>>>(
      x, Wv, v_ws, BL, KVN, D_MODEL);

  rope_rms<<<dim3((BL * NUM_HEADS) / 8), blk, 0, stream>>>(
      q_ws, cs, sn, NUM_HEADS, BL * NUM_HEADS);
  rope_rms<<<dim3((BL * NUM_KV) / 8), blk, 0, stream>>>(
      k_ws, cs, sn, NUM_KV, BL * NUM_KV);

  flash_attn<<<dim3(BATCH * NUM_HEADS * (SEQ / 16)), dim3(32), 0, stream>>>(
      q_ws, k_ws, v_ws, a_ws);

  gemm_f32<<<dim3(BL / 128, D_MODEL / 128), blk, 0, stream>>>(
      a_ws, Wo, out, BL, D_MODEL, D_MODEL);
}